// DilatedAttention_91018946937254
// MI455X (gfx1250) — compile-verified
//
#include <hip/hip_runtime.h>

// CDNA5 / gfx1250 dilated attention, single pass, fp32 WMMA.
// One wave32 handles one 16-position "super-window" (16 rows x d=64).
// S = Q*K^T (16x16) via V_WMMA_F32_16X16X4_F32; three masked softmaxes
// (w=4/8/16 dilation patterns are sub-blocks of S) done with DPP16 lane-xor
// butterflies (no LDS traffic); combined prob matrix P = sum_g w_g*softmax_g;
// O = P*V via WMMA again (P staged C-layout -> A-layout through LDS).

typedef float v2f __attribute__((ext_vector_type(2)));
typedef float v8f __attribute__((ext_vector_type(8)));

#define WPB 8           // waves per block (256 threads)
#define LDS_STRIDE 20   // 16 + pad, avoids half-vs-half bank conflicts

// DPP16 ctrl codes
#define DPP_XOR1 0xB1   // quad_perm(1,0,3,2)
#define DPP_XOR2 0x4E   // quad_perm(2,3,0,1)
#define DPP_XOR4 0x164  // row_xmask:4
#define DPP_XOR8 0x168  // row_xmask:8

template <int CTRL>
__device__ __forceinline__ float xorf(float x) {
    return __int_as_float(
        __builtin_amdgcn_update_dpp(0, __float_as_int(x), CTRL, 0xF, 0xF, true));
}

__global__ __launch_bounds__(256) void dilattn_wmma_kernel(
    const float* __restrict__ q,
    const float* __restrict__ k,
    const float* __restrict__ v,
    const float* __restrict__ alpha,
    float* __restrict__ out,
    int n_sw)
{
    __shared__ float ldsP[WPB * 16 * LDS_STRIDE];

    const int tid  = (int)threadIdx.x;
    const int lane = tid & 31;
    const int wIB  = tid >> 5;
    const int W    = (int)blockIdx.x * WPB + wIB;
    if (W >= n_sw) return;                 // wave-uniform

    const int r    = lane & 15;            // row (A) / col (B,C) index
    const int half = lane >> 4;            // 0 or 1

    const size_t base = (size_t)W * 16 * 64;

    // ---- mixing weights = softmax(alpha) over 3 groups ----
    const float a0 = alpha[0], a1 = alpha[1], a2 = alpha[2];
    const float am = fmaxf(a0, fmaxf(a1, a2));
    const float e0 = __expf(a0 - am), e1 = __expf(a1 - am), e2 = __expf(a2 - am);
    const float winv = __builtin_amdgcn_rcpf(e0 + e1 + e2);
    const float w0 = e0 * winv, w1 = e1 * winv, w2 = e2 * winv;

    // ---- S = Q * K^T (16x16), 16 accumulating f32 WMMA steps over K=64 ----
    // A layout (16x4 f32): lane r holds row r; VGPR0 = col 4kk+2*half, VGPR1 = +1
    // B layout (4x16 f32): lane r holds col N=r; VGPR0 = K^T row 4kk+2*half = K[r][...]
    v8f S = {0.f,0.f,0.f,0.f,0.f,0.f,0.f,0.f};
    const float* qrow = q + base + (size_t)r * 64 + 2 * half;
    const float* krow = k + base + (size_t)r * 64 + 2 * half;
    #pragma unroll
    for (int kk = 0; kk < 16; ++kk) {
        v2f a = *(const v2f*)(qrow + 4 * kk);
        v2f b = *(const v2f*)(krow + 4 * kk);
        S = __builtin_amdgcn_wmma_f32_16x16x4_f32(false, a, false, b,
                                                  (short)0, S, false, false);
    }

    // ---- masked softmaxes; element (vgpr m, lane): i = m + 8*half, j = r ----
    const float scale = 0.125f;  // 64^-0.5
    const float NEGINF = -__builtin_inff();
    float P[8];
    #pragma unroll
    for (int m = 0; m < 8; ++m) P[m] = 0.f;

    #pragma unroll
    for (int m = 0; m < 8; ++m) {
        const int i = m + 8 * half;
        const float sv = S[m] * scale;

        // group 0: w=4, every row; j in same 4-block as i; reduce over xor{1,2}
        {
            const bool msk = ((i >> 2) == (r >> 2));
            float s = msk ? sv : NEGINF;
            float mx = fmaxf(s, xorf<DPP_XOR1>(s));
            mx = fmaxf(mx, xorf<DPP_XOR2>(mx));
            float e = msk ? __expf(s - mx) : 0.f;
            float sum = e + xorf<DPP_XOR1>(e);
            sum += xorf<DPP_XOR2>(sum);
            P[m] += msk ? e * (w0 * __builtin_amdgcn_rcpf(sum)) : 0.f;
        }
        // group 1: w=8,r=2: i,j even, same 8-block; reduce over xor{2,4}
        {
            const bool msk = (((i & 1) == 0) && ((r & 1) == 0) &&
                              ((i >> 3) == (r >> 3)));
            float s = msk ? sv : NEGINF;
            float mx = fmaxf(s, xorf<DPP_XOR2>(s));
            mx = fmaxf(mx, xorf<DPP_XOR4>(mx));
            float e = msk ? __expf(s - mx) : 0.f;
            float sum = e + xorf<DPP_XOR2>(e);
            sum += xorf<DPP_XOR4>(sum);
            P[m] += msk ? e * (w1 * __builtin_amdgcn_rcpf(sum)) : 0.f;
        }
        // group 2: w=16,r=4: i,j multiples of 4; reduce over xor{4,8}
        {
            const bool msk = (((i & 3) == 0) && ((r & 3) == 0));
            float s = msk ? sv : NEGINF;
            float mx = fmaxf(s, xorf<DPP_XOR4>(s));
            mx = fmaxf(mx, xorf<DPP_XOR8>(mx));
            float e = msk ? __expf(s - mx) : 0.f;
            float sum = e + xorf<DPP_XOR4>(e);
            sum += xorf<DPP_XOR8>(sum);
            P[m] += msk ? e * (w2 * __builtin_amdgcn_rcpf(sum)) : 0.f;
        }
    }

    // ---- stage P (C-layout) -> LDS -> reload in A-layout ----
    float* myLds = ldsP + wIB * 16 * LDS_STRIDE;
    #pragma unroll
    for (int m = 0; m < 8; ++m) {
        const int i = m + 8 * half;
        myLds[i * LDS_STRIDE + r] = P[m];
    }
    asm volatile("s_wait_dscnt 0" ::: "memory");   // intra-wave LDS RAW fence

    // ---- O = P (16x16) * V (16x64): 4 d-chunks x 4 K-steps of WMMA ----
    const float* vb = v + base;
    float* ob = out + base;
    #pragma unroll
    for (int nc = 0; nc < 4; ++nc) {
        v8f O = {0.f,0.f,0.f,0.f,0.f,0.f,0.f,0.f};
        #pragma unroll
        for (int kk = 0; kk < 4; ++kk) {
            v2f a = *(const v2f*)(myLds + r * LDS_STRIDE + 4 * kk + 2 * half);
            v2f b;
            b.x = vb[(size_t)(4 * kk + 2 * half)     * 64 + nc * 16 + r];
            b.y = vb[(size_t)(4 * kk + 2 * half + 1) * 64 + nc * 16 + r];
            O = __builtin_amdgcn_wmma_f32_16x16x4_f32(false, a, false, b,
                                                      (short)0, O, false, false);
        }
        #pragma unroll
        for (int m = 0; m < 8; ++m) {
            const int i = m + 8 * half;
            ob[(size_t)i * 64 + nc * 16 + r] = O[m];
        }
    }
}

extern "C" void kernel_launch(void* const* d_in, const int* in_sizes, int n_in,
                              void* d_out, int out_size, void* d_ws, size_t ws_size,
                              hipStream_t stream) {
    const float* q     = (const float*)d_in[0];
    const float* k     = (const float*)d_in[1];
    const float* v     = (const float*)d_in[2];
    const float* alpha = (const float*)d_in[3];
    float* out = (float*)d_out;

    const int n_elems = in_sizes[0];          // b*h*s*d = 16777216
    const int n_sw = n_elems / (16 * 64);     // super-windows of 16 positions
    const int blocks = (n_sw + WPB - 1) / WPB;

    dilattn_wmma_kernel<<<blocks, 256, 0, stream>>>(q, k, v, alpha, out, n_sw);
}